// MLP_50560355008545
// MI455X (gfx1250) — compile-verified
//
#include <hip/hip_runtime.h>
#include <hip/hip_bf16.h>
#include <math.h>
#include <float.h>

typedef __attribute__((ext_vector_type(2))) float v2f;
typedef __attribute__((ext_vector_type(8))) float v8f;

#define BB 256
#define SS 512
#define HH 768

// ---------------------------------------------------------------------------
// Kernel 1: compute span bounds from the 0/1 id vectors, then max-pool ONLY
// the rows inside the span (<=32 rows instead of 512 -> 16x less HBM traffic),
// writing the concatenated [pooled | span_max] row of shape [B, 2H].
// ---------------------------------------------------------------------------
__global__ __launch_bounds__(256)
void spanmax_cat_kernel(const float* __restrict__ se,
                        const float* __restrict__ pooled,
                        const int* __restrict__ left_ids,
                        const int* __restrict__ t_ids,
                        float* __restrict__ cat)
{
    __shared__ int s_left, s_t;
    const int b = blockIdx.x;
    const int t = threadIdx.x;
    if (t == 0) { s_left = 0; s_t = 0; }
    __syncthreads();
    if (left_ids[b * 256 + t] != 0) atomicAdd(&s_left, 1);
    if (t < 64 && t_ids[b * 64 + t] != 0) atomicAdd(&s_t, 1);
    __syncthreads();

    int start = s_left - 1;            // left_len
    int len   = s_t - 2;               // target_len
    int end   = start + len;
    if (start < 0) start = 0;
    if (end > SS) end = SS;

    const float* base = se + (size_t)b * SS * HH;
    float* crow = cat + (size_t)b * (2 * HH);
    for (int h = t; h < HH; h += 256) {
        float m = -FLT_MAX;                       // matches jnp.finfo.min fill
        for (int s = start; s < end; ++s)
            m = fmaxf(m, base[(size_t)s * HH + h]);
        crow[HH + h] = m;
        crow[h]      = pooled[(size_t)b * HH + h];
    }
}

// ---------------------------------------------------------------------------
// Kernel 2: fp32 WMMA GEMM  out = act(A[MxK] @ W[KxN] + bias)
// Block = 128 threads = 4 waves. Block tile: 16 rows x 64 cols; wave w owns
// the 16x16 tile at columns n0 = 64*bx + 16*w. A chunk (16x32) staged in LDS.
// Fragment layouts per CDNA5 ISA (V_WMMA_F32_16X16X4_F32):
//   A 16x4: lane L: M = L%16; VGPR0/1 hold K = {2*(L/16), 2*(L/16)+1}
//   B 4x16: lane L: N = L%16; VGPR0/1 hold K = {2*(L/16), 2*(L/16)+1}
//   C/D   : lane L: N = L%16; VGPR r -> M = r + 8*(L/16)
// ---------------------------------------------------------------------------
template<bool TANH>
__global__ __launch_bounds__(128)
void gemm_bias_act_wmma(const float* __restrict__ A,
                        const float* __restrict__ W,
                        const float* __restrict__ bias,
                        float* __restrict__ out,
                        int M, int N, int K)
{
    __shared__ float sA[16][33];                 // pad 33: conflict-free columns

    const int wave = threadIdx.x >> 5;           // 0..3
    const int lane = threadIdx.x & 31;
    const int half = lane >> 4;                  // 0 | 1
    const int l16  = lane & 15;
    const int m0   = blockIdx.y * 16;
    const int n0   = blockIdx.x * 64 + wave * 16;

    v8f acc = {};

    for (int k0 = 0; k0 < K; k0 += 32) {
        // cooperative A-chunk load: 16x32 floats, coalesced along K
        for (int i = threadIdx.x; i < 16 * 32; i += 128) {
            int r = i >> 5, c = i & 31;
            sA[r][c] = A[(size_t)(m0 + r) * K + k0 + c];
        }
        __syncthreads();

#pragma unroll
        for (int kk = 0; kk < 32; kk += 4) {
            const int ak = kk + half * 2;        // this lane's K pair
            v2f a, bfrag;
            a.x = sA[l16][ak];
            a.y = sA[l16][ak + 1];
            const float* wp = W + (size_t)(k0 + ak) * N + n0 + l16;
            bfrag.x = wp[0];                     // row k, 16 contiguous lanes
            bfrag.y = wp[N];                     // row k+1
            acc = __builtin_amdgcn_wmma_f32_16x16x4_f32(
                      /*neg_a=*/false, a, /*neg_b=*/false, bfrag,
                      /*c_mod=*/(short)0, acc,
                      /*reuse_a=*/false, /*reuse_b=*/false);
        }
        __syncthreads();
    }

    const int n = n0 + l16;
    const float bn = bias[n];
#pragma unroll
    for (int r = 0; r < 8; ++r) {
        const int m = m0 + r + 8 * half;
        float v = acc[r] + bn;
        if (TANH) v = tanhf(v);
        out[(size_t)m * N + n] = v;
    }
}

// ---------------------------------------------------------------------------
// Kernel 3: tiny head  logits[b,o] = h3[b,:384] . W4[:,o] + b4[o]   (OUT=3)
// ---------------------------------------------------------------------------
__global__ __launch_bounds__(128)
void final_fc_kernel(const float* __restrict__ h3,
                     const float* __restrict__ W4,
                     const float* __restrict__ b4,
                     float* __restrict__ out)
{
    const int idx = blockIdx.x * 128 + threadIdx.x;
    if (idx >= BB * 3) return;
    const int b = idx / 3, o = idx % 3;
    const float* hp = h3 + (size_t)b * 384;
    float acc = b4[o];
#pragma unroll 4
    for (int k = 0; k < 384; ++k)
        acc += hp[k] * W4[k * 3 + o];
    out[idx] = acc;
}

// ---------------------------------------------------------------------------
extern "C" void kernel_launch(void* const* d_in, const int* in_sizes, int n_in,
                              void* d_out, int out_size, void* d_ws, size_t ws_size,
                              hipStream_t stream) {
    const float* se       = (const float*)d_in[0];   // [256,512,768]
    const float* pooled   = (const float*)d_in[1];   // [256,768]
    const int*   left_ids = (const int*)  d_in[2];   // [256,256]
    const int*   t_ids    = (const int*)  d_in[3];   // [256,64]
    const float* W1 = (const float*)d_in[4];         // [1536,768]
    const float* b1 = (const float*)d_in[5];
    const float* W2 = (const float*)d_in[6];         // [768,768]
    const float* b2 = (const float*)d_in[7];
    const float* W3 = (const float*)d_in[8];         // [768,384]
    const float* b3 = (const float*)d_in[9];
    const float* W4 = (const float*)d_in[10];        // [384,3]
    const float* b4 = (const float*)d_in[11];

    float* ws  = (float*)d_ws;
    float* cat = ws;                                 // 256*1536
    float* h1  = cat + (size_t)BB * 2 * HH;          // 256*768
    float* h2  = h1  + (size_t)BB * HH;              // 256*768
    float* h3  = h2  + (size_t)BB * HH;              // 256*384

    spanmax_cat_kernel<<<BB, 256, 0, stream>>>(se, pooled, left_ids, t_ids, cat);

    gemm_bias_act_wmma<true><<<dim3(HH / 64, BB / 16), 128, 0, stream>>>(
        cat, W1, b1, h1, BB, HH, 2 * HH);            // 256 x 768 <- K=1536
    gemm_bias_act_wmma<true><<<dim3(HH / 64, BB / 16), 128, 0, stream>>>(
        h1, W2, b2, h2, BB, HH, HH);                 // 256 x 768 <- K=768
    gemm_bias_act_wmma<true><<<dim3((HH / 2) / 64, BB / 16), 128, 0, stream>>>(
        h2, W3, b3, h3, BB, HH / 2, HH);             // 256 x 384 <- K=768

    final_fc_kernel<<<(BB * 3 + 127) / 128, 128, 0, stream>>>(h3, W4, b4, (float*)d_out);
}